// RGCL_Loss_2259152797808
// MI455X (gfx1250) — compile-verified
//
#include <hip/hip_runtime.h>
#include <hip/hip_bf16.h>

// ---------------- constants from the reference ----------------
#define GAMMA      0.8f
#define EPSV       1e-10f
#define RHO_I      8.0f
#define RHO_T      8.0f
#define ETA_INIT   1e-05f
#define BETA_U     0.5f
#define GRAD_CLIP  5.0f
#define TAU_MIN    0.005f
#define TAU_MAX    0.05f

typedef float v2f __attribute__((ext_vector_type(2)));
typedef float v8f __attribute__((ext_vector_type(8)));

// ------- CDNA5 async global->LDS copy (ASYNCcnt-tracked), via inline asm -------
__device__ __forceinline__ void async_ld_b128(unsigned lds_off, const float* g)
{
    asm volatile("global_load_async_to_lds_b128 %0, %1, off"
                 :: "v"(lds_off), "v"(g) : "memory");
}
__device__ __forceinline__ void wait_async0()
{
    asm volatile("s_wait_asynccnt 0" ::: "memory");
}

// =====================================================================
// GEMM: C[NxN] = A[NxD] * B[NxD]^T with V_WMMA_F32_16X16X4_F32.
// 256 threads = 8 waves arranged 4(row) x 2(col); each wave owns a 16x32
// output (two independent accumulators) -> workgroup tile 64x64.
// K staged through double-buffered LDS panels filled with
// GLOBAL_LOAD_ASYNC_TO_LDS_B128 (overlaps copy of panel k+1 with WMMA on k).
// =====================================================================
#define TILE_K 32
#define ROWPAD 4            // 16B pad: keeps B128 chunks aligned, skews banks

__global__ __launch_bounds__(256)
void rgcl_gemm_nt(const float* __restrict__ A, const float* __restrict__ Bm,
                  float* __restrict__ C, int N, int D)
{
    __shared__ float As[2][64][TILE_K + ROWPAD];
    __shared__ float Bs[2][64][TILE_K + ROWPAD];

    const int tid     = threadIdx.x;       // 0..255
    const int wave    = tid >> 5;          // wave32
    const int lane    = tid & 31;
    const int waveRow = wave >> 1;         // 0..3
    const int waveCol = wave & 1;          // 0..1
    const int half    = lane >> 4;         // 0|1 (half-wave)
    const int l16     = lane & 15;

    const int rowBase = blockIdx.y * 64;
    const int colBase = blockIdx.x * 64;

    const int aRow  = waveRow * 16 + l16;
    const int bRow0 = waveCol * 32 + l16;
    const int bRow1 = bRow0 + 16;

    v8f acc0 = {0.f, 0.f, 0.f, 0.f, 0.f, 0.f, 0.f, 0.f};
    v8f acc1 = {0.f, 0.f, 0.f, 0.f, 0.f, 0.f, 0.f, 0.f};

    // stage one 64x32 A panel + 64x32 B panel into buffer `buf`
    auto issue = [&](int buf, int k0) {
#pragma unroll
        for (int i = 0; i < 2; ++i) {
            int c  = tid + i * 256;        // 512 16B chunks per panel
            int r  = c >> 3;               // 8 chunks per 32-float row
            int k4 = (c & 7) << 2;
            async_ld_b128((unsigned)(size_t)&As[buf][r][k4],
                          &A[(size_t)(rowBase + r) * D + k0 + k4]);
            async_ld_b128((unsigned)(size_t)&Bs[buf][r][k4],
                          &Bm[(size_t)(colBase + r) * D + k0 + k4]);
        }
    };

    issue(0, 0);
    wait_async0();
    __syncthreads();

    int buf = 0;
    for (int k0 = 0; k0 < D; k0 += TILE_K) {
        if (k0 + TILE_K < D)
            issue(buf ^ 1, k0 + TILE_K);   // overlap next panel with compute

#pragma unroll
        for (int kk = 0; kk < TILE_K; kk += 4) {
            // A frag (16x4 f32): lanes 0-15 -> K={kk,kk+1}, lanes 16-31 -> K={kk+2,kk+3}
            const int ka = kk + 2 * half;
            v2f a, b0, b1;
            a.x  = As[buf][aRow][ka];
            a.y  = As[buf][aRow][ka + 1];
            b0.x = Bs[buf][bRow0][ka];
            b0.y = Bs[buf][bRow0][ka + 1];
            b1.x = Bs[buf][bRow1][ka];
            b1.y = Bs[buf][bRow1][ka + 1];
            acc0 = __builtin_amdgcn_wmma_f32_16x16x4_f32(
                false, a, false, b0, (short)0, acc0, false, false);
            acc1 = __builtin_amdgcn_wmma_f32_16x16x4_f32(
                false, a, false, b1, (short)0, acc1, false, false);
        }
        wait_async0();
        __syncthreads();
        buf ^= 1;
    }

    // C layout: vgpr r, lanes 0-15 -> M=r; lanes 16-31 -> M=r+8; N=l16
    const int crow  = rowBase + waveRow * 16 + 8 * half;
    const int ccol0 = colBase + waveCol * 32 + l16;
#pragma unroll
    for (int r = 0; r < 8; ++r) {
        C[(size_t)(crow + r) * N + ccol0]      = acc0[r];
        C[(size_t)(crow + r) * N + ccol0 + 16] = acc1[r];
    }
}

// =====================================================================
// Per-row LSE pass. One 256-thread block per row of M (row-major NxN).
// Image side uses M = sim; text side uses M = sim^T (columns -> rows).
// =====================================================================
__global__ __launch_bounds__(256)
void rgcl_row_pass(const float* __restrict__ M, int N,
                   const int*   __restrict__ ids,
                   const float* __restrict__ s_in,
                   const float* __restrict__ b_in,
                   const float* __restrict__ u_in,
                   const float* __restrict__ tau_in,
                   const int*   __restrict__ epoch_p,
                   float rho,
                   float* __restrict__ out_s, float* __restrict__ out_b,
                   float* __restrict__ out_u, float* __restrict__ out_tau,
                   float* __restrict__ ws_loss, float* __restrict__ ws_tw,
                   float* __restrict__ ws_tauold)
{
    __shared__ float r0[256], r1[256], r2[256];

    const int row = blockIdx.x;
    const int tid = threadIdx.x;
    const int id  = ids[row];

    const float tau     = tau_in[id];
    const float inv_tau = 1.0f / tau;
    const float old_b   = b_in[id];
    const float diag    = M[(size_t)row * N + row];
    const float* __restrict__ Mrow = M + (size_t)row * N;

    // ---- phase 1: running max of idt over the row (diagonal -> 0) ----
    float m = -3.402823466e38f;
    for (int j = tid; j < N; j += 256)
        m = fmaxf(m, (Mrow[j] - diag) * inv_tau);
    r0[tid] = m;
    __syncthreads();
#pragma unroll
    for (int s = 128; s > 0; s >>= 1) {
        if (tid < s) r0[tid] = fmaxf(r0[tid], r0[tid + s]);
        __syncthreads();
    }
    const float b_row = fmaxf(r0[0], old_b);
    __syncthreads();

    // ---- phase 2: masked exp sums: sum e, sum e*d, sum e*x ----
    float se = 0.f, sd = 0.f, sx = 0.f;
    for (int j = tid; j < N; j += 256) {
        if (j == row) continue;
        const float d = Mrow[j] - diag;
        const float x = d * inv_tau;
        const float e = expf(x - b_row);
        se += e; sd += e * d; sx += e * x;
    }
    r0[tid] = se; r1[tid] = sd; r2[tid] = sx;
    __syncthreads();
#pragma unroll
    for (int s = 128; s > 0; s >>= 1) {
        if (tid < s) {
            r0[tid] += r0[tid + s];
            r1[tid] += r1[tid + s];
            r2[tid] += r2[tid + s];
        }
        __syncthreads();
    }

    if (tid == 0) {
        const float g = r0[0];
        float sI;
        if (*epoch_p == 0) sI = g;
        else sI = (1.0f - GAMMA) * s_in[id] * expf(old_b - b_row) + GAMMA * g;
        const float inv = 1.0f / (sI + EPSV);
        const float loss = r1[0] * inv;
        float tw = logf(sI / (float)(N - 1)) + b_row + rho - r2[0] * inv;
        tw = fminf(fmaxf(tw, -GRAD_CLIP), GRAD_CLIP);
        const float u = (1.0f - BETA_U) * u_in[id] + BETA_U * tw;
        const float tau_new = fminf(fmaxf(tau - ETA_INIT * u, TAU_MIN), TAU_MAX);

        out_s[id]   = sI;
        out_b[id]   = b_row;
        out_u[id]   = u;
        out_tau[id] = tau_new;
        ws_loss[row]   = loss;
        ws_tw[row]     = tw;
        ws_tauold[row] = tau;
    }
}

// ---------------- bulk copy of state arrays into d_out ----------------
__global__ void rgcl_copy_f32(float* __restrict__ dst,
                              const float* __restrict__ src, int n)
{
    for (int i = blockIdx.x * blockDim.x + threadIdx.x; i < n;
         i += gridDim.x * blockDim.x)
        dst[i] = src[i];
}

// ---------------- deterministic scalar means ----------------
__global__ __launch_bounds__(256)
void rgcl_finalize(const float* __restrict__ li, const float* __restrict__ lt,
                   const float* __restrict__ ti, const float* __restrict__ tt,
                   const float* __restrict__ wi, const float* __restrict__ wt,
                   int n, float* __restrict__ out)
{
    __shared__ float red[256];
    const float* arrs[6] = {li, lt, ti, tt, wi, wt};
    float sums[6];
#pragma unroll
    for (int a = 0; a < 6; ++a) {
        float s = 0.f;
        for (int j = threadIdx.x; j < n; j += 256) s += arrs[a][j];
        red[threadIdx.x] = s;
        __syncthreads();
        for (int st = 128; st > 0; st >>= 1) {
            if (threadIdx.x < st) red[threadIdx.x] += red[threadIdx.x + st];
            __syncthreads();
        }
        sums[a] = red[0];
        __syncthreads();
    }
    if (threadIdx.x == 0) {
        const float invn = 1.0f / (float)n;
        out[0] = (sums[0] + sums[1]) * invn;  // total_loss
        out[1] = sums[2] * invn;              // tau_img.mean
        out[2] = sums[3] * invn;              // tau_txt.mean
        out[3] = sums[4] * invn;              // tw_i.mean
        out[4] = sums[5] * invn;              // tw_t.mean
    }
}

extern "C" void kernel_launch(void* const* d_in, const int* in_sizes, int n_in,
                              void* d_out, int out_size, void* d_ws, size_t ws_size,
                              hipStream_t stream)
{
    const float* img     = (const float*)d_in[0];
    const float* txt     = (const float*)d_in[1];
    const float* s_I     = (const float*)d_in[2];
    const float* s_T     = (const float*)d_in[3];
    const float* b_I     = (const float*)d_in[4];
    const float* b_T     = (const float*)d_in[5];
    const float* u_I     = (const float*)d_in[6];
    const float* u_T     = (const float*)d_in[7];
    const float* tau_I   = (const float*)d_in[8];
    const float* tau_T   = (const float*)d_in[9];
    const int* image_ids = (const int*)d_in[10];
    const int* text_ids  = (const int*)d_in[11];
    const int* epoch     = (const int*)d_in[12];

    const int nI = in_sizes[2];
    const int nT = in_sizes[3];
    const int Bn = in_sizes[10];            // 4096
    const int D  = in_sizes[0] / Bn;        // 512

    float* out    = (float*)d_out;
    float* o_sI   = out + 5;
    float* o_sT   = o_sI + nI;
    float* o_bI   = o_sT + nT;
    float* o_bT   = o_bI + nI;
    float* o_uI   = o_bT + nT;
    float* o_uT   = o_uI + nI;
    float* o_tauI = o_uT + nT;
    float* o_tauT = o_tauI + nI;

    float* ws = (float*)d_ws;
    const size_t NN = (size_t)Bn * (size_t)Bn;
    float* sim      = ws;                  // 64 MB  (L2-resident)
    float* simT     = ws + NN;             // 64 MB  (L2-resident)
    float* w_loss_i = simT + NN;
    float* w_tw_i   = w_loss_i + Bn;
    float* w_tau_i  = w_tw_i + Bn;
    float* w_loss_t = w_tau_i + Bn;
    float* w_tw_t   = w_loss_t + Bn;
    float* w_tau_t  = w_tw_t + Bn;

    // 1) bulk-copy state arrays into the output (scatter overwrites later)
    const int cb = 1024;
    rgcl_copy_f32<<<cb, 256, 0, stream>>>(o_sI,   s_I,   nI);
    rgcl_copy_f32<<<cb, 256, 0, stream>>>(o_sT,   s_T,   nT);
    rgcl_copy_f32<<<cb, 256, 0, stream>>>(o_bI,   b_I,   nI);
    rgcl_copy_f32<<<cb, 256, 0, stream>>>(o_bT,   b_T,   nT);
    rgcl_copy_f32<<<cb, 256, 0, stream>>>(o_uI,   u_I,   nI);
    rgcl_copy_f32<<<cb, 256, 0, stream>>>(o_uT,   u_T,   nT);
    rgcl_copy_f32<<<cb, 256, 0, stream>>>(o_tauI, tau_I, nI);
    rgcl_copy_f32<<<cb, 256, 0, stream>>>(o_tauT, tau_T, nT);

    // 2) sim = img @ txt^T ; simT = txt @ img^T  (FP32 WMMA; both live in L2)
    dim3 ggrid(Bn / 64, Bn / 64);
    rgcl_gemm_nt<<<ggrid, 256, 0, stream>>>(img, txt, sim,  Bn, D);
    rgcl_gemm_nt<<<ggrid, 256, 0, stream>>>(txt, img, simT, Bn, D);

    // 3) image rows / text columns (rows of simT), fully coalesced
    rgcl_row_pass<<<Bn, 256, 0, stream>>>(sim,  Bn, image_ids, s_I, b_I, u_I, tau_I,
                                          epoch, RHO_I, o_sI, o_bI, o_uI, o_tauI,
                                          w_loss_i, w_tw_i, w_tau_i);
    rgcl_row_pass<<<Bn, 256, 0, stream>>>(simT, Bn, text_ids,  s_T, b_T, u_T, tau_T,
                                          epoch, RHO_T, o_sT, o_bT, o_uT, o_tauT,
                                          w_loss_t, w_tw_t, w_tau_t);

    // 4) deterministic means -> out[0..4]
    rgcl_finalize<<<1, 256, 0, stream>>>(w_loss_i, w_loss_t, w_tau_i, w_tau_t,
                                         w_tw_i, w_tw_t, Bn, out);
}